// IPLS_82781199663554
// MI455X (gfx1250) — compile-verified
//
#include <hip/hip_runtime.h>
#include <hip/hip_bf16.h>

typedef float v2f __attribute__((ext_vector_type(2)));
typedef float v8f __attribute__((ext_vector_type(8)));

#define N_FEAT 262144
#define N_TGT  8192
#define N_LAT  32
#define EPS_F  1e-7f

// ---------------- workspace layout (float offsets) ----------------
#define F_CHUNK   4096
#define F_NCHUNK  (N_FEAT / F_CHUNK)      // 64
#define F_TILES   5                        // 80 padded rows / 16
#define F_NPAIR   15                       // tiles*(tiles+1)/2
#define T_CHUNK   2048
#define T_NCHUNK  (N_TGT / T_CHUNK)        // 4
#define T_TILES   3                        // 48 padded rows / 16
#define T_NPAIR   6
#define ZROW_LEN  4352u                    // >= F_CHUNK + 16, multiple of 256

#define WS_XC   0u
#define WS_YC   (WS_XC + N_FEAT)
#define WS_PF   (WS_YC + N_TGT)                         // 15*64*256
#define WS_PT   (WS_PF + F_NPAIR * F_NCHUNK * 256u)     // 6*4*256
#define WS_GF   (WS_PT + T_NPAIR * T_NCHUNK * 256u)     // 80*80
#define WS_GT   (WS_GF + 80u * 80u)                     // 48*48
#define WS_T2   (WS_GT + 48u * 48u)
#define WS_U1   (WS_T2 + N_LAT)
#define WS_KY   (WS_U1 + N_LAT)
#define WS_ZROW (WS_KY + N_LAT)                         // ZROW_LEN zeros

// ---------------- output layout (float offsets) ----------------
#define O_MUX 0u
#define O_MUY (O_MUX + N_FEAT)
#define O_U   (O_MUY + N_TGT)
#define O_WZ  (O_U + N_LAT)
#define O_CZ  (O_WZ + (unsigned)N_LAT * N_FEAT)
#define O_TSS (O_CZ + (unsigned)N_LAT * N_TGT)
#define O_BZ  (O_TSS + N_LAT)
#define O_P   (O_BZ + N_LAT)

// ================= zero-fill for the padded-row source =================
__global__ void k_fillzero(float* __restrict__ dst, int len) {
  int i = blockIdx.x * blockDim.x + threadIdx.x;
  if (i < len) dst[i] = 0.f;
}

// ================= kernel 1: running mean + centering =================
__global__ void k_center(const float* __restrict__ xin,
                         const float* __restrict__ mu_in,
                         const int* __restrict__ n,
                         float* __restrict__ mu_out,
                         float* __restrict__ xc, int len) {
  int i = blockIdx.x * blockDim.x + threadIdx.x;
  if (i >= len) return;
  float nn = (float)(*n + 1);                    // n_new
  float mu = (mu_in[i] * nn + xin[i]) / (nn + 1.0f);
  mu_out[i] = mu;
  xc[i] = xin[i] - mu;
}

// ================= kernel 2: Gram matrix via WMMA f32 16x16x4 =================
__device__ __forceinline__ const float* gram_row_ptr(int g, const float* r0,
                                                     const float* m1,
                                                     const float* m2, int N) {
  if (g == 0) return r0;
  if (g <= N_LAT) return m1 + (size_t)(g - 1) * (size_t)N;
  return m2 + (size_t)(g - 1 - N_LAT) * (size_t)N;
}

// One wave per block. blockIdx.x = upper-triangular tile pair, blockIdx.y = K chunk.
// Rows: 0 = r0 (xc/yc), 1..32 = m1 rows, 33..64 = m2 rows, rest read a zero row.
// A (16x4 f32) fragment layout per ISA: lane l holds row base+(l&15),
// K offset 2*(l>>4), 2 consecutive floats -> float2 load. B (4x16) layout is the
// mirror of A for V*V^T, so the same per-lane loader serves both operands.
// Padded rows are handled by pointing those lanes at a zero-filled row, so the
// hot loop is pure load -> wmma with no exec-mask selects.
__global__ void k_gram(const float* __restrict__ r0, const float* __restrict__ m1,
                       const float* __restrict__ m2, int R, int N,
                       int chunk_cols, int nchunks, int T,
                       const float* __restrict__ zrow,
                       float* __restrict__ partials) {
  const int pairIdx = blockIdx.x;
  const int chunk = blockIdx.y;
  int I = 0, p = pairIdx;
  while (p >= T - I) { p -= T - I; ++I; }
  const int J = I + p;

  const int lane = threadIdx.x;
  const int sub = lane & 15;
  const int half = lane >> 4;

  const int ga = I * 16 + sub;
  const int gb = J * 16 + sub;

  const size_t coff = (size_t)chunk * (size_t)chunk_cols + (size_t)(2 * half);
  const float* pa = (ga < R) ? (gram_row_ptr(ga, r0, m1, m2, N) + coff)
                             : (zrow + (size_t)(2 * half));
  const float* pb = (gb < R) ? (gram_row_ptr(gb, r0, m1, m2, N) + coff)
                             : (zrow + (size_t)(2 * half));

  v8f acc0 = {0.f, 0.f, 0.f, 0.f, 0.f, 0.f, 0.f, 0.f};
  v8f acc1 = acc0, acc2 = acc0, acc3 = acc0;

  for (int k = 0; k < chunk_cols; k += 16) {
    v2f a0 = *(const v2f*)(pa + k + 0);
    v2f a1 = *(const v2f*)(pa + k + 4);
    v2f a2 = *(const v2f*)(pa + k + 8);
    v2f a3 = *(const v2f*)(pa + k + 12);
    v2f b0 = *(const v2f*)(pb + k + 0);
    v2f b1 = *(const v2f*)(pb + k + 4);
    v2f b2 = *(const v2f*)(pb + k + 8);
    v2f b3 = *(const v2f*)(pb + k + 12);
    acc0 = __builtin_amdgcn_wmma_f32_16x16x4_f32(false, a0, false, b0, (short)0, acc0, false, false);
    acc1 = __builtin_amdgcn_wmma_f32_16x16x4_f32(false, a1, false, b1, (short)0, acc1, false, false);
    acc2 = __builtin_amdgcn_wmma_f32_16x16x4_f32(false, a2, false, b2, (short)0, acc2, false, false);
    acc3 = __builtin_amdgcn_wmma_f32_16x16x4_f32(false, a3, false, b3, (short)0, acc3, false, false);
  }
  v8f acc = (acc0 + acc1) + (acc2 + acc3);
  float* po = partials + ((size_t)pairIdx * (size_t)nchunks + (size_t)chunk) * 256u;
#pragma unroll
  for (int v = 0; v < 8; ++v) po[v * 32 + lane] = acc[v];
}

// Fixed-order chunk reduction + scatter into full symmetric Gram matrix.
// C/D layout per ISA: VGPR v, lane l -> M = v + 8*(l>>4), N = l&15.
__global__ void k_gram_reduce(const float* __restrict__ partials, int nchunks,
                              int T, int GDIM, float* __restrict__ G) {
  const int pairIdx = blockIdx.x;
  const int t = threadIdx.x;  // 0..255
  int I = 0, p = pairIdx;
  while (p >= T - I) { p -= T - I; ++I; }
  const int J = I + p;
  float s = 0.f;
  for (int c = 0; c < nchunks; ++c)
    s += partials[((size_t)pairIdx * (size_t)nchunks + (size_t)c) * 256u + t];
  const int v = t >> 5, lane = t & 31;
  const int M = v + 8 * (lane >> 4), Nc = lane & 15;
  const int gr = I * 16 + M, gc = J * 16 + Nc;
  G[gr * GDIM + gc] = s;
  if (I != J) G[gc * GDIM + gr] = s;
}

// ================= kernel 3: serial component scan on Gram scalars =================
__global__ void k_scan(const float* __restrict__ Gf, const float* __restrict__ Gt,
                       const float* __restrict__ u_in, const float* __restrict__ tss_in,
                       const float* __restrict__ bz_in, const int* __restrict__ n,
                       float* __restrict__ t2o, float* __restrict__ u1o,
                       float* __restrict__ kyo, float* __restrict__ u_out,
                       float* __restrict__ tss_out, float* __restrict__ bz_out) {
  if (threadIdx.x != 0 || blockIdx.x != 0) return;
  const int GF = 80, GT = 48;
  const bool burn = (*n + 1) <= 3;  // BURN_IN
  if (burn) {
    // wz = Wz_i + u_i*xc ; tz = xc@wz/(||wz||+eps) ; only Wz, tss change.
    float qx = Gf[0];  // ||xc||^2
    for (int i = 0; i < N_LAT; ++i) {
      float u0 = u_in[i];
      float aw = Gf[0 * GF + (1 + i)];        // dot(xc, Wz_i)
      float sw = Gf[(1 + i) * GF + (1 + i)];  // ||Wz_i||^2
      float tz = (aw + u0 * qx) / (sqrtf(sw + 2.f * u0 * aw + u0 * u0 * qx) + EPS_F);
      tss_out[i] = tss_in[i] + tz * tz;
      u_out[i] = u0;
      bz_out[i] = bz_in[i];
      t2o[i] = 0.f; u1o[i] = u0; kyo[i] = 0.f;  // makes output kernels copy P/Cz
    }
    return;
  }
  // xr = xcoef[0]*xc + sum_j xcoef[1+j]*P_j ; yr likewise over {yc, Cz_j}.
  float xcoef[33], ycoef[33];
  for (int b = 0; b < 33; ++b) { xcoef[b] = 0.f; ycoef[b] = 0.f; }
  xcoef[0] = 1.f; ycoef[0] = 1.f;
  float q = Gf[0];  // ||xr||^2
  float r = Gt[0];  // ||yr||^2
  for (int i = 0; i < N_LAT; ++i) {
    const int colW = 1 + i, colP = 33 + i, colC = 1 + i;
    float a = 0.f, pdot = 0.f, c = 0.f;
    for (int b = 0; b < 33; ++b) {
      int gr = (b == 0) ? 0 : (32 + b);  // basis row: xc or P_{b-1}
      a    += xcoef[b] * Gf[gr * GF + colW];
      pdot += xcoef[b] * Gf[gr * GF + colP];
      int gt = (b == 0) ? 0 : b;         // basis row: yc or Cz_{b-1}
      c    += ycoef[b] * Gt[gt * GT + colC];
    }
    const float s  = Gf[colW * GF + colW];   // ||Wz_i||^2
    const float pp = Gf[colP * GF + colP];   // ||P_i||^2
    const float sc = Gt[colC * GT + colC];   // ||Cz_i||^2
    const float tssi = tss_in[i];
    // inner power iteration 1 (uses u0)
    float u0v = u_in[i];
    float tz1 = (a + u0v * q) / (sqrtf(s + 2.f * u0v * a + u0v * u0v * q) + EPS_F);
    float tss1 = tssi + tz1 * tz1;
    float t1 = tz1 / sqrtf(tss1);
    float ncz1 = sqrtf(sc + 2.f * t1 * c + t1 * t1 * r);
    float u1 = (c + t1 * r) / ncz1;
    // inner power iteration 2 (uses u1) -> final wz/cz/P use u1, t2
    float tz2 = (a + u1 * q) / (sqrtf(s + 2.f * u1 * a + u1 * u1 * q) + EPS_F);
    float tss2 = tssi + tz2 * tz2;
    float t2 = tz2 / sqrtf(tss2);
    float ncz2 = sqrtf(sc + 2.f * t2 * c + t2 * t2 * r);
    float u2 = (c + t2 * r) / ncz2;
    float bzn = bz_in[i] + u2 * tz2;
    float bb = bzn / sqrtf(tss2);
    float ky = bb * t2 / ncz2;  // yr2 = yr - ky*cz
    t2o[i] = t2; u1o[i] = u1; kyo[i] = ky;
    u_out[i] = u2; tss_out[i] = tss2; bz_out[i] = bzn;
    // deflation in coefficient space: xr' = (1-t2^2)xr - t2*P_i
    float f1 = 1.f - t2 * t2;
    for (int b = 0; b < 33; ++b) xcoef[b] *= f1;
    xcoef[1 + i] = -t2;
    // yr' = (1 - ky*t2)yr - ky*Cz_i
    float f2 = 1.f - ky * t2;
    for (int b = 0; b < 33; ++b) ycoef[b] *= f2;
    ycoef[1 + i] = -ky;
    // norm recurrences
    q = f1 * f1 * q - 2.f * t2 * f1 * pdot + t2 * t2 * pp;
    r = f2 * f2 * r - 2.f * ky * f2 * c + ky * ky * sc;
  }
}

// ================= kernel 4: feature-side outputs (Wz, P) =================
__global__ void k_feat_out(const float* __restrict__ xc, const float* __restrict__ Wz,
                           const float* __restrict__ P, const float* __restrict__ t2,
                           const float* __restrict__ u1, float* __restrict__ WzOut,
                           float* __restrict__ POut) {
  int f = blockIdx.x * blockDim.x + threadIdx.x;
  if (f >= N_FEAT) return;
  float xr = xc[f];
#pragma unroll 4
  for (int i = 0; i < N_LAT; ++i) {
    size_t idx = (size_t)i * N_FEAT + f;
    float w = Wz[idx] + u1[i] * xr;   // wz from last inner iteration
    float po = P[idx] + t2[i] * xr;   // P_new
    WzOut[idx] = w;
    POut[idx] = po;
    xr = xr - t2[i] * po;             // deflate x
  }
}

// ================= kernel 5: target-side outputs (Cz) =================
__global__ void k_tgt_out(const float* __restrict__ yc, const float* __restrict__ Cz,
                          const float* __restrict__ t2, const float* __restrict__ ky,
                          float* __restrict__ CzOut) {
  int g = blockIdx.x * blockDim.x + threadIdx.x;
  if (g >= N_TGT) return;
  float yr = yc[g];
#pragma unroll 4
  for (int i = 0; i < N_LAT; ++i) {
    size_t idx = (size_t)i * N_TGT + g;
    float cz = Cz[idx] + t2[i] * yr;  // cz from last inner iteration
    CzOut[idx] = cz;
    yr = yr - ky[i] * cz;             // yr - b*t*C
  }
}

extern "C" void kernel_launch(void* const* d_in, const int* in_sizes, int n_in,
                              void* d_out, int out_size, void* d_ws, size_t ws_size,
                              hipStream_t stream) {
  const float* x    = (const float*)d_in[0];
  const float* y    = (const float*)d_in[1];
  const float* mu_x = (const float*)d_in[2];
  const float* mu_y = (const float*)d_in[3];
  const float* u    = (const float*)d_in[4];
  const float* Wz   = (const float*)d_in[5];
  const float* Cz   = (const float*)d_in[6];
  const float* tss  = (const float*)d_in[7];
  const float* bz   = (const float*)d_in[8];
  const float* P    = (const float*)d_in[9];
  const int*   n    = (const int*)d_in[10];
  float* out = (float*)d_out;
  float* ws = (float*)d_ws;

  // 0) zero row for padded fragment lanes (rewritten every call)
  k_fillzero<<<ZROW_LEN / 256, 256, 0, stream>>>(ws + WS_ZROW, (int)ZROW_LEN);

  // 1) mean update + centering
  k_center<<<N_FEAT / 256, 256, 0, stream>>>(x, mu_x, n, out + O_MUX, ws + WS_XC, N_FEAT);
  k_center<<<N_TGT / 256, 256, 0, stream>>>(y, mu_y, n, out + O_MUY, ws + WS_YC, N_TGT);

  // 2) Gram matrices via WMMA f32 16x16x4 (one wave per block)
  dim3 gf(F_NPAIR, F_NCHUNK);
  k_gram<<<gf, 32, 0, stream>>>(ws + WS_XC, Wz, P, 65, N_FEAT, F_CHUNK, F_NCHUNK,
                                F_TILES, ws + WS_ZROW, ws + WS_PF);
  dim3 gt(T_NPAIR, T_NCHUNK);
  k_gram<<<gt, 32, 0, stream>>>(ws + WS_YC, Cz, nullptr, 33, N_TGT, T_CHUNK, T_NCHUNK,
                                T_TILES, ws + WS_ZROW, ws + WS_PT);
  k_gram_reduce<<<F_NPAIR, 256, 0, stream>>>(ws + WS_PF, F_NCHUNK, F_TILES, 80, ws + WS_GF);
  k_gram_reduce<<<T_NPAIR, 256, 0, stream>>>(ws + WS_PT, T_NCHUNK, T_TILES, 48, ws + WS_GT);

  // 3) serial scalar scan over 32 components (tiny)
  k_scan<<<1, 32, 0, stream>>>(ws + WS_GF, ws + WS_GT, u, tss, bz, n,
                               ws + WS_T2, ws + WS_U1, ws + WS_KY,
                               out + O_U, out + O_TSS, out + O_BZ);

  // 4) big elementwise output passes
  k_feat_out<<<N_FEAT / 256, 256, 0, stream>>>(ws + WS_XC, Wz, P, ws + WS_T2,
                                               ws + WS_U1, out + O_WZ, out + O_P);
  k_tgt_out<<<N_TGT / 256, 256, 0, stream>>>(ws + WS_YC, Cz, ws + WS_T2, ws + WS_KY,
                                             out + O_CZ);
}